// ADBSLoss_8847632630276
// MI455X (gfx1250) — compile-verified
//
#include <hip/hip_runtime.h>
#include <hip/hip_bf16.h>
#include <math.h>

typedef __attribute__((ext_vector_type(2))) float v2f;
typedef __attribute__((ext_vector_type(8))) float v8f;

#define ALPHA_C   0.05f
#define B_ROWS    16384
#define C_CLS     4096
#define D_DIM     768
#define NT2       (C_CLS / 32)   // 128 super-tile blocks (32x32) per dimension

// ---------------- diag[i] = P[i] . P[i] ----------------
__global__ void adbs_diag_kernel(const float* __restrict__ P,
                                 float* __restrict__ diag) {
    __shared__ float red[256];
    const int row = blockIdx.x;
    const float* p = P + (size_t)row * D_DIM;
    float s = 0.f;
    for (int t = threadIdx.x; t < D_DIM; t += 256) {
        float v = p[t];
        s += v * v;
    }
    red[threadIdx.x] = s;
    __syncthreads();
    for (int off = 128; off > 0; off >>= 1) {
        if (threadIdx.x < off) red[threadIdx.x] += red[threadIdx.x + off];
        __syncthreads();
    }
    if (threadIdx.x == 0) diag[row] = red[0];
}

// ---------------- per-row NLL (cross entropy), b128 loads ----------------
__global__ void adbs_ce_kernel(const float* __restrict__ logits,
                               const int* __restrict__ targets,
                               float* __restrict__ ce_partial) {
    __shared__ float red[256];
    const int row = blockIdx.x;
    const float* x = logits + (size_t)row * C_CLS;
    const float4* x4 = (const float4*)x;

    // pass 1: row max (coalesced 16B/lane)
    float m = -INFINITY;
    for (int t = threadIdx.x; t < C_CLS / 4; t += 256) {
        float4 v = x4[t];
        m = fmaxf(m, fmaxf(fmaxf(v.x, v.y), fmaxf(v.z, v.w)));
    }
    red[threadIdx.x] = m;
    __syncthreads();
    for (int off = 128; off > 0; off >>= 1) {
        if (threadIdx.x < off)
            red[threadIdx.x] = fmaxf(red[threadIdx.x], red[threadIdx.x + off]);
        __syncthreads();
    }
    const float rowmax = red[0];
    __syncthreads();

    // pass 2: sum exp (row is cache-resident from pass 1)
    float s = 0.f;
    for (int t = threadIdx.x; t < C_CLS / 4; t += 256) {
        float4 v = x4[t];
        s += expf(v.x - rowmax) + expf(v.y - rowmax) +
             expf(v.z - rowmax) + expf(v.w - rowmax);
    }
    red[threadIdx.x] = s;
    __syncthreads();
    for (int off = 128; off > 0; off >>= 1) {
        if (threadIdx.x < off) red[threadIdx.x] += red[threadIdx.x + off];
        __syncthreads();
    }
    if (threadIdx.x == 0) {
        int t = targets[row];
        // nll = -(x_t - rowmax - log(sumexp))
        ce_partial[row] = rowmax + logf(red[0]) - x[t];
    }
}

// ---------------- IC 32x32 super-tiles via V_WMMA_F32_16X16X4_F32 ----------
// One wave per 32x32 super-tile of G = P P^T, register-blocked 2x2:
// 4 loads -> 4 WMMAs per K-step (1 load/WMMA, 8 FLOP/B).  Upper triangle
// only; off-diagonal super-tiles evaluate both orientations via G symmetry.
__global__ void adbs_ic_tile_kernel(const float* __restrict__ P,
                                    const float* __restrict__ bnd,
                                    const float* __restrict__ diag,
                                    float* __restrict__ ic_partial) {
    const int it = blockIdx.x;       // row super-tile
    const int jt = blockIdx.y;       // col super-tile
    const int pidx = jt * NT2 + it;
    if (jt < it) {                   // lower triangle: emit zero partial
        if (threadIdx.x == 0) ic_partial[pidx] = 0.f;
        return;
    }

    const int lane = threadIdx.x & 31;
    const int half = lane >> 4;      // 0: K pair {0,1}, 1: K pair {2,3}
    const int lm   = lane & 15;

    // Fragment rows (A layout == B layout for P P^T tiles).
    const float* arow0 = P + (size_t)(it * 32 + lm) * D_DIM + 2 * half;
    const float* arow1 = arow0 + (size_t)16 * D_DIM;
    const float* brow0 = P + (size_t)(jt * 32 + lm) * D_DIM + 2 * half;
    const float* brow1 = brow0 + (size_t)16 * D_DIM;

    v8f acc00 = {}, acc01 = {}, acc10 = {}, acc11 = {};
    for (int k = 0; k < D_DIM; k += 4) {
        v2f a0 = *(const v2f*)(arow0 + k);
        v2f a1 = *(const v2f*)(arow1 + k);
        v2f b0 = *(const v2f*)(brow0 + k);
        v2f b1 = *(const v2f*)(brow1 + k);
        acc00 = __builtin_amdgcn_wmma_f32_16x16x4_f32(false, a0, false, b0,
                                                      (short)0, acc00, false, false);
        acc01 = __builtin_amdgcn_wmma_f32_16x16x4_f32(false, a0, false, b1,
                                                      (short)0, acc01, false, false);
        acc10 = __builtin_amdgcn_wmma_f32_16x16x4_f32(false, a1, false, b0,
                                                      (short)0, acc10, false, false);
        acc11 = __builtin_amdgcn_wmma_f32_16x16x4_f32(false, a1, false, b1,
                                                      (short)0, acc11, false, false);
    }

    // C/D layout: VGPR r, lanes 0-15 -> M=r, lanes 16-31 -> M=r+8; N = lane%16.
    const bool mirror = (it != jt);
    float s = 0.f;
#pragma unroll
    for (int ib = 0; ib < 2; ++ib) {
#pragma unroll
        for (int jb = 0; jb < 2; ++jb) {
            const v8f acc = (ib == 0) ? ((jb == 0) ? acc00 : acc01)
                                      : ((jb == 0) ? acc10 : acc11);
            const int j = jt * 32 + jb * 16 + lm;   // col index, fixed per lane
            const float bj = bnd[j];
            const float dj = diag[j];
#pragma unroll
            for (int r = 0; r < 8; ++r) {
                const int i = it * 32 + ib * 16 + r + 8 * half;
                const float g  = acc[r];
                const float bi = bnd[i];
                const float di = diag[i];
                const float cf = (1.f - bi) * di + (bj - 1.f) * g;  // pair (i,j)
                if (i != j) s += fmaxf(cf, 0.f);
                if (mirror) {                                       // pair (j,i)
                    const float cm = (1.f - bj) * dj + (bi - 1.f) * g;
                    s += fmaxf(cm, 0.f);
                }
            }
        }
    }
    // wave32 reduction
    for (int off = 16; off > 0; off >>= 1) s += __shfl_xor(s, off, 32);
    if (lane == 0) ic_partial[pidx] = s;
}

// ---------------- deterministic final reduction ----------------
__global__ void adbs_finalize_kernel(const float* __restrict__ ce_partial,
                                     const float* __restrict__ ic_partial,
                                     float* __restrict__ out) {
    __shared__ float red[256];
    const int tid = threadIdx.x;

    float s = 0.f;
    for (int i = tid; i < B_ROWS; i += 256) s += ce_partial[i];
    red[tid] = s;
    __syncthreads();
    for (int off = 128; off > 0; off >>= 1) {
        if (tid < off) red[tid] += red[tid + off];
        __syncthreads();
    }
    const float cls = red[0] / (float)B_ROWS;
    __syncthreads();

    s = 0.f;
    for (int i = tid; i < NT2 * NT2; i += 256) s += ic_partial[i];
    red[tid] = s;
    __syncthreads();
    for (int off = 128; off > 0; off >>= 1) {
        if (tid < off) red[tid] += red[tid + off];
        __syncthreads();
    }
    const float ic = red[0] / ((float)C_CLS * (float)(C_CLS - 1));

    if (tid == 0) {
        out[0] = cls + ALPHA_C * ic;   // total
        out[1] = cls;
        out[2] = ic;
    }
}

extern "C" void kernel_launch(void* const* d_in, const int* in_sizes, int n_in,
                              void* d_out, int out_size, void* d_ws, size_t ws_size,
                              hipStream_t stream) {
    const float* logits     = (const float*)d_in[0];
    const int*   targets    = (const int*)d_in[1];
    const float* prototypes = (const float*)d_in[2];
    const float* boundaries = (const float*)d_in[3];
    float* out = (float*)d_out;

    float* ws_f    = (float*)d_ws;
    float* diag    = ws_f;                       // 4096
    float* ce_part = ws_f + C_CLS;               // 16384
    float* ic_part = ws_f + C_CLS + B_ROWS;      // 16384

    adbs_diag_kernel<<<C_CLS, 256, 0, stream>>>(prototypes, diag);
    adbs_ce_kernel<<<B_ROWS, 256, 0, stream>>>(logits, targets, ce_part);
    adbs_ic_tile_kernel<<<dim3(NT2, NT2), 32, 0, stream>>>(prototypes, boundaries,
                                                           diag, ic_part);
    adbs_finalize_kernel<<<1, 256, 0, stream>>>(ce_part, ic_part, out);
}